// HandcraftedKeyModel_19000935318236
// MI455X (gfx1250) — compile-verified
//
#include <hip/hip_runtime.h>
#include <hip/hip_bf16.h>

typedef __attribute__((ext_vector_type(2))) float v2f;
typedef __attribute__((ext_vector_type(8))) float v8f;

#define T_LEN    524288
#define CHUNK    256
#define NCHUNK   (T_LEN / CHUNK)     // 2048
#define GROUP    64
#define NGROUP   (NCHUNK / GROUP)    // 32
#define WPB      8                   // waves per block (256 threads, wave32)
#define DECAY_F  0.9995001249791693f
#define MAXH     5.0f

// ---------------------------------------------------------------------------
// Phase A: per-chunk capped-affine summaries (a,b,c) per key.
//   step map f(h) = min(d*h + inc, 5); compose new-after-old:
//   a' = a*d ; b' = b*d + inc ; c' = min(c*d + inc, 5)
// One wave per chunk, lane = key (12 active). Notes fetched 16 at a time;
// per-key step bitmask built via shuffles (off the serial fma/min chain).
// ---------------------------------------------------------------------------
__global__ __launch_bounds__(256) void ka_summaries(const int* __restrict__ x,
                                                    float* __restrict__ sum) {
  const int lane  = threadIdx.x & 31;
  const int wave  = threadIdx.x >> 5;
  const int chunk = blockIdx.x * WPB + wave;
  const int t0c   = chunk * CHUNK;
  const int key   = lane;

  float a = 1.0f, b = 0.0f, c = 3.0e38f;
  for (int bt = 0; bt < CHUNK; bt += 16) {
    int nm = x[t0c + bt + (lane & 15)] % 12;   // lanes 16..31 duplicate, harmless
    unsigned mask = 0u;
#pragma unroll
    for (int s = 0; s < 16; ++s) {
      int ks = __shfl(nm, s, 32);
      mask |= (ks == key) ? (1u << s) : 0u;
    }
    if (lane < 12) {
#pragma unroll
      for (int s = 0; s < 16; ++s) {
        float inc = ((mask >> s) & 1u) ? 1.0f : 0.0f;
        c = fminf(fmaf(c, DECAY_F, inc), MAXH);
        b = fmaf(b, DECAY_F, inc);
        a = a * DECAY_F;
      }
    }
  }
  if (lane < 12) {
    sum[chunk * 36 +      key] = a;
    sum[chunk * 36 + 12 + key] = b;
    sum[chunk * 36 + 24 + key] = c;
  }
}

// ---------------------------------------------------------------------------
// Phase B: exact starting h for every chunk. Two-level prefix:
//   stage1: wave g composes its 64 chunk summaries -> group summary (LDS)
//   stage2: wave 0 serial-scans 32 group summaries -> group start h
//   stage3: wave g replays its 64 summaries from the group start -> h0[chunk]
// ---------------------------------------------------------------------------
__global__ __launch_bounds__(1024) void kb_prefix(const float* __restrict__ sum,
                                                  float* __restrict__ h0) {
  __shared__ float gsA[NGROUP][12], gsB[NGROUP][12], gsC[NGROUP][12];
  __shared__ float gstart[NGROUP][12];
  const int lane = threadIdx.x & 31;
  const int g    = threadIdx.x >> 5;   // group id, 0..31
  const int key  = lane;

  if (lane < 12) {
    float a = 1.0f, b = 0.0f, c = 3.0e38f;
    for (int i = 0; i < GROUP; ++i) {
      const int ch = g * GROUP + i;
      float ai = sum[ch * 36 +      key];
      float bi = sum[ch * 36 + 12 + key];
      float ci = sum[ch * 36 + 24 + key];
      c = fminf(fmaf(ai, c, bi), ci);
      b = fmaf(ai, b, bi);
      a = a * ai;
    }
    gsA[g][key] = a; gsB[g][key] = b; gsC[g][key] = c;
  }
  __syncthreads();
  if (threadIdx.x < 12) {                    // wave 0 lanes 0..11
    float h = 0.0f;
    for (int gg = 0; gg < NGROUP; ++gg) {
      gstart[gg][key] = h;
      h = fminf(fmaf(gsA[gg][key], h, gsB[gg][key]), gsC[gg][key]);
    }
  }
  __syncthreads();
  if (lane < 12) {
    float h = gstart[g][key];
    for (int i = 0; i < GROUP; ++i) {
      const int ch = g * GROUP + i;
      h0[ch * 12 + key] = h;
      h = fminf(fmaf(sum[ch * 36 + key], h, sum[ch * 36 + 12 + key]),
                sum[ch * 36 + 24 + key]);
    }
  }
}

// ---------------------------------------------------------------------------
// Phase C: replay scan from exact h0, heats tile (16x12) -> LDS, then
// WMMA F32 16x16x4 (3 K-steps = K12, 6 N-tiles = 96 padded cols),
// store [16,84] fp32 rows. Output traffic (176 MB) is the only real cost.
// ---------------------------------------------------------------------------
__global__ __launch_bounds__(256) void kc_heats_matmul(const int* __restrict__ x,
                                                       const float* __restrict__ W,
                                                       const float* __restrict__ h0,
                                                       float* __restrict__ out) {
  __shared__ float tile[WPB][16 * 16];   // per-wave 16 rows x 16 (12 used) heats
  const int lane  = threadIdx.x & 31;
  const int wave  = threadIdx.x >> 5;
  const int chunk = blockIdx.x * WPB + wave;
  const int tbase = chunk * CHUNK;
  const int key   = lane;
  float* wtile = tile[wave];

  // B fragments: B is KxN (4x16). lanes 0-15: N=lane, VGPR0=K(kb+0), VGPR1=K(kb+1);
  // lanes 16-31: N=lane-16, K(kb+2)/K(kb+3). W is [12][84] row-major; pad N to 96.
  v2f bf[6][3];
  {
    const int coln = lane & 15;
    const int kh   = (lane < 16) ? 0 : 2;
#pragma unroll
    for (int nt = 0; nt < 6; ++nt) {
      const int n = nt * 16 + coln;
#pragma unroll
      for (int j = 0; j < 3; ++j) {
        const int k0 = 4 * j + kh;
        float w0 = 0.0f, w1 = 0.0f;
        if (n < 84) { w0 = W[k0 * 84 + n]; w1 = W[(k0 + 1) * 84 + n]; }
        bf[nt][j] = (v2f){w0, w1};
      }
    }
  }

  float h = 0.0f;
  if (lane < 12) h = h0[chunk * 12 + key];

  const int row  = lane & 15;            // A-matrix row (M) for this lane
  const int koff = (lane < 16) ? 0 : 2;  // K offset within a K=4 slab

  for (int bt = 0; bt < CHUNK; bt += 16) {
    const int t0 = tbase + bt;
    int nm = x[t0 + (lane & 15)] % 12;
    unsigned mask = 0u;
#pragma unroll
    for (int s = 0; s < 16; ++s) {
      int ks = __shfl(nm, s, 32);
      mask |= (ks == key) ? (1u << s) : 0u;
    }
    if (lane < 12) {
#pragma unroll
      for (int s = 0; s < 16; ++s) {
        float inc = ((mask >> s) & 1u) ? 1.0f : 0.0f;
        h = fminf(fmaf(h, DECAY_F, inc), MAXH);
        wtile[s * 16 + key] = h;
      }
    }
    __syncthreads();

    // A fragments (16x4 f32): lane<16 -> M=lane, K=kb+0/kb+1; lane>=16 -> M=lane-16, K=kb+2/kb+3
    v2f a0 = (v2f){wtile[row * 16 + 0 + koff], wtile[row * 16 + 1 + koff]};
    v2f a1 = (v2f){wtile[row * 16 + 4 + koff], wtile[row * 16 + 5 + koff]};
    v2f a2 = (v2f){wtile[row * 16 + 8 + koff], wtile[row * 16 + 9 + koff]};
    __syncthreads();

#pragma unroll
    for (int nt = 0; nt < 6; ++nt) {
      v8f acc = (v8f){0.f, 0.f, 0.f, 0.f, 0.f, 0.f, 0.f, 0.f};
      acc = __builtin_amdgcn_wmma_f32_16x16x4_f32(false, a0, false, bf[nt][0], (short)0, acc, false, false);
      acc = __builtin_amdgcn_wmma_f32_16x16x4_f32(false, a1, false, bf[nt][1], (short)0, acc, false, false);
      acc = __builtin_amdgcn_wmma_f32_16x16x4_f32(false, a2, false, bf[nt][2], (short)0, acc, false, false);

      const int n = nt * 16 + (lane & 15);
      if (n < 84) {
        const int rbase = t0 + ((lane < 16) ? 0 : 8);   // D: VGPR r -> rows r / r+8
#pragma unroll
        for (int r = 0; r < 8; ++r) {
          out[(size_t)(rbase + r) * 84 + n] = acc[r];
        }
      }
    }
  }
}

extern "C" void kernel_launch(void* const* d_in, const int* in_sizes, int n_in,
                              void* d_out, int out_size, void* d_ws, size_t ws_size,
                              hipStream_t stream) {
  const int*   x  = (const int*)d_in[0];    // [1, 524288] int32 notes
  const float* W  = (const float*)d_in[1];  // [12, 84] fp32
  float* out = (float*)d_out;               // [1, T, 7, 12] fp32 == [T, 84]
  float* sum = (float*)d_ws;                // [NCHUNK][3][12] summaries
  float* h0  = sum + NCHUNK * 36;           // [NCHUNK][12] chunk-start heats

  ka_summaries   <<<NCHUNK / WPB, 256, 0, stream>>>(x, sum);
  kb_prefix      <<<1, 1024, 0, stream>>>(sum, h0);
  kc_heats_matmul<<<NCHUNK / WPB, 256, 0, stream>>>(x, W, h0, out);
}